// BrainInflate_6459630813500
// MI455X (gfx1250) — compile-verified
//
#include <hip/hip_runtime.h>

typedef float v2f __attribute__((ext_vector_type(2)));
typedef float v8f __attribute__((ext_vector_type(8)));

#define NB 32           // batches per hemisphere
#define NM 64           // total meshes (2 hemis * 32)
#define STEPF 0.1f
#define EPSF 1e-8f

__device__ inline v8f wmma4(v2f a, v2f b, v8f c) {
    // D = A(16x4 f32) * B(4x16 f32) + C(16x16 f32)
    return __builtin_amdgcn_wmma_f32_16x16x4_f32(false, a, false, b, (short)0, c, false, false);
}

// ---------------------------------------------------------------- zero fill
__global__ void k_zero(float* __restrict__ p, long n) {
    long i = (long)blockIdx.x * blockDim.x + threadIdx.x;
    long stride = (long)gridDim.x * blockDim.x;
    for (; i < n; i += stride) p[i] = 0.f;
}

// ------------------------------------------------------- vertex degree count
__global__ void k_count_faces(const int* __restrict__ faces, float* __restrict__ cnt, int F) {
    int f = blockIdx.x * blockDim.x + threadIdx.x;
    if (f >= F) return;
    atomicAdd(&cnt[faces[3 * f + 0]], 2.f);
    atomicAdd(&cnt[faces[3 * f + 1]], 2.f);
    atomicAdd(&cnt[faces[3 * f + 2]], 2.f);
}

// ------------------------------------------------ rcnt = 1/cnt (topology-const)
__global__ void k_recip(const float* __restrict__ cnt, float* __restrict__ rcnt, int V) {
    int i = blockIdx.x * blockDim.x + threadIdx.x;
    if (i < V) rcnt[i] = 1.f / cnt[i];
}

// --------------------------------------------- build concatenated weight mats
// Wc1 (24x16): rows 0-8 = W1s, rows 12-20 = W1n, rest zero.
// Wc2 (32x16): rows 0-15 = W2s, rows 16-31 = W2n.
// Wo16 (16x16): cols 0-2 = Wo (16x3), rest zero.
__global__ void k_prep_weights(const float* __restrict__ W1s, const float* __restrict__ W1n,
                               const float* __restrict__ W2s, const float* __restrict__ W2n,
                               const float* __restrict__ Wo,
                               float* __restrict__ Wc1, float* __restrict__ Wc2,
                               float* __restrict__ Wo16) {
    int t = threadIdx.x;  // 512 threads
    if (t < 384) {
        int r = t >> 4, n = t & 15;
        float v = 0.f;
        if (r < 9) v = W1s[r * 16 + n];
        else if (r >= 12 && r < 21) v = W1n[(r - 12) * 16 + n];
        Wc1[t] = v;
    }
    {
        int r = t >> 4, n = t & 15;
        Wc2[t] = (r < 16) ? W2s[r * 16 + n] : W2n[(r - 16) * 16 + n];
    }
    if (t < 256) {
        int r = t >> 4, n = t & 15;
        Wo16[t] = (n < 3) ? Wo[r * 3 + n] : 0.f;
    }
}

// --------------------------- per-mesh bbox normalize, init vcur / sulc / size
__global__ void k_prep_mesh(const float* __restrict__ vlh, const float* __restrict__ vrh,
                            float* __restrict__ vcur, float* __restrict__ sulc,
                            float* __restrict__ cs, int V, int Vp) {
    __shared__ float smn[3][256], smx[3][256];
    int m = blockIdx.x, t = threadIdx.x;
    const float* src = (m < NB) ? (vlh + (long)m * V * 3) : (vrh + (long)(m - NB) * V * 3);
    float mn[3] = {3.4e38f, 3.4e38f, 3.4e38f};
    float mx[3] = {-3.4e38f, -3.4e38f, -3.4e38f};
    for (int i = t; i < V; i += 256) {
        #pragma unroll
        for (int c = 0; c < 3; ++c) {
            float x = src[3 * i + c];
            mn[c] = fminf(mn[c], x);
            mx[c] = fmaxf(mx[c], x);
        }
    }
    #pragma unroll
    for (int c = 0; c < 3; ++c) { smn[c][t] = mn[c]; smx[c][t] = mx[c]; }
    __syncthreads();
    for (int s = 128; s > 0; s >>= 1) {
        if (t < s) {
            #pragma unroll
            for (int c = 0; c < 3; ++c) {
                smn[c][t] = fminf(smn[c][t], smn[c][t + s]);
                smx[c][t] = fmaxf(smx[c][t], smx[c][t + s]);
            }
        }
        __syncthreads();
    }
    float cen[3], siz[3];
    #pragma unroll
    for (int c = 0; c < 3; ++c) {
        cen[c] = 0.5f * (smn[c][0] + smx[c][0]);
        siz[c] = smx[c][0] - cen[c];
    }
    if (t == 0) { cs[m * 4 + 0] = siz[0]; cs[m * 4 + 1] = siz[1]; cs[m * 4 + 2] = siz[2]; }
    for (int i = t; i < V; i += 256) {
        long idx = (long)m * Vp + i;
        #pragma unroll
        for (int c = 0; c < 3; ++c) vcur[idx * 3 + c] = (src[3 * i + c] - cen[c]) / siz[c];
        sulc[idx] = 0.f;
    }
}

// ----------------------- face normals (scatter) + cot-Laplacian (scatter)
__global__ void k_face_geom(const int* __restrict__ faces, const float* __restrict__ vcur,
                            float* __restrict__ nrm, float* __restrict__ lap, int Vp, int F) {
    long tid = (long)blockIdx.x * blockDim.x + threadIdx.x;
    long total = (long)NM * F;
    if (tid >= total) return;
    int m = (int)(tid / F), f = (int)(tid % F);
    int i0 = faces[3 * f], i1 = faces[3 * f + 1], i2 = faces[3 * f + 2];
    const float* vb = vcur + (long)m * (long)Vp * 3;
    float p0[3], p1[3], p2[3];
    #pragma unroll
    for (int c = 0; c < 3; ++c) { p0[c] = vb[3 * i0 + c]; p1[c] = vb[3 * i1 + c]; p2[c] = vb[3 * i2 + c]; }
    float e1[3], e2[3];
    #pragma unroll
    for (int c = 0; c < 3; ++c) { e1[c] = p1[c] - p0[c]; e2[c] = p2[c] - p0[c]; }
    float fn[3];
    fn[0] = e1[1] * e2[2] - e1[2] * e2[1];
    fn[1] = e1[2] * e2[0] - e1[0] * e2[2];
    fn[2] = e1[0] * e2[1] - e1[1] * e2[0];
    float inva = 1.f / (sqrtf(fn[0] * fn[0] + fn[1] * fn[1] + fn[2] * fn[2]) + EPSF);
    float sgn = (m >= NB) ? -1.f : 1.f;  // reversed winding on rh hemisphere
    long r0 = ((long)m * Vp + i0) * 3, r1 = ((long)m * Vp + i1) * 3, r2 = ((long)m * Vp + i2) * 3;
    #pragma unroll
    for (int c = 0; c < 3; ++c) {
        atomicAdd(&nrm[r0 + c], sgn * fn[c]);
        atomicAdd(&nrm[r1 + c], sgn * fn[c]);
        atomicAdd(&nrm[r2 + c], sgn * fn[c]);
    }
    // cotangents (all three cross magnitudes equal 2*area)
    float a1[3], b1v[3], a2[3], b2v[3];
    #pragma unroll
    for (int c = 0; c < 3; ++c) {
        a1[c] = p2[c] - p1[c]; b1v[c] = p0[c] - p1[c];
        a2[c] = p0[c] - p2[c]; b2v[c] = p1[c] - p2[c];
    }
    float c0 = (e1[0] * e2[0] + e1[1] * e2[1] + e1[2] * e2[2]) * inva;
    float c1 = (a1[0] * b1v[0] + a1[1] * b1v[1] + a1[2] * b1v[2]) * inva;
    float c2 = (a2[0] * b2v[0] + a2[1] * b2v[1] + a2[2] * b2v[2]) * inva;
    #pragma unroll
    for (int c = 0; c < 3; ++c) {
        float g0 = c1 * (p2[c] - p0[c]) + c2 * (p1[c] - p0[c]);
        float g1 = c0 * (p2[c] - p1[c]) + c2 * (p0[c] - p1[c]);
        float g2 = c0 * (p1[c] - p2[c]) + c1 * (p0[c] - p2[c]);
        atomicAdd(&lap[r0 + c], g0);
        atomicAdd(&lap[r1 + c], g1);
        atomicAdd(&lap[r2 + c], g2);
    }
}

// ----------------------- normalize normals, assemble feats = [v, n, 0.5*lap]
__global__ void k_vert_feats(const float* __restrict__ vcur, float* __restrict__ nrm,
                             const float* __restrict__ lap, float* __restrict__ X1,
                             int V, int Vp) {
    long tid = (long)blockIdx.x * blockDim.x + threadIdx.x;
    long total = (long)NM * V;
    if (tid >= total) return;
    int m = (int)(tid / V), i = (int)(tid % V);
    long idx = (long)m * Vp + i;
    float nx = nrm[idx * 3 + 0], ny = nrm[idx * 3 + 1], nz = nrm[idx * 3 + 2];
    float inv = 1.f / (sqrtf(nx * nx + ny * ny + nz * nz) + EPSF);
    nx *= inv; ny *= inv; nz *= inv;
    nrm[idx * 3 + 0] = nx; nrm[idx * 3 + 1] = ny; nrm[idx * 3 + 2] = nz;
    float* xr = X1 + idx * 24;
    xr[0] = vcur[idx * 3 + 0]; xr[1] = vcur[idx * 3 + 1]; xr[2] = vcur[idx * 3 + 2];
    xr[3] = nx; xr[4] = ny; xr[5] = nz;
    xr[6] = 0.5f * lap[idx * 3 + 0];
    xr[7] = 0.5f * lap[idx * 3 + 1];
    xr[8] = 0.5f * lap[idx * 3 + 2];
    // cols 9..23 stay zero / are filled by neighbor scatter
}

// -------- neighbor-sum scatter: X[:,DSTOFF+j] += sum over face-neighbors X[:,j]
template <int NCOLS, int NFEAT, int DSTOFF>
__global__ void k_nbr_scatter(const int* __restrict__ faces, float* __restrict__ X,
                              int Vp, int F) {
    long tid = (long)blockIdx.x * blockDim.x + threadIdx.x;
    long total = (long)NM * F;
    if (tid >= total) return;
    int m = (int)(tid / F), f = (int)(tid % F);
    int i0 = faces[3 * f], i1 = faces[3 * f + 1], i2 = faces[3 * f + 2];
    float* X0 = X + ((long)m * Vp + i0) * NCOLS;
    float* X1r = X + ((long)m * Vp + i1) * NCOLS;
    float* X2r = X + ((long)m * Vp + i2) * NCOLS;
    float a[NFEAT], b[NFEAT], c[NFEAT];
    #pragma unroll
    for (int j = 0; j < NFEAT; ++j) { a[j] = X0[j]; b[j] = X1r[j]; c[j] = X2r[j]; }
    #pragma unroll
    for (int j = 0; j < NFEAT; ++j) {
        atomicAdd(&X0[DSTOFF + j], b[j] + c[j]);
        atomicAdd(&X1r[DSTOFF + j], a[j] + c[j]);
        atomicAdd(&X2r[DSTOFF + j], a[j] + b[j]);
    }
}

// ------------------- MLP layer 1: X2[:,0:16] = relu(X1(24) @ Wc1(24x16) + b1)
__global__ void k_mlp1(const float* __restrict__ X1, const float* __restrict__ Wc1,
                       const float* __restrict__ b1, const float* __restrict__ rcnt,
                       float* __restrict__ X2, int V, int Vp) {
    int m = blockIdx.y;
    int wid = threadIdx.x >> 5;
    int lane = threadIdx.x & 31;
    int tile = blockIdx.x * 8 + wid;        // tile < Vp/16 by construction
    int n = lane & 15;
    int half = lane >> 4;
    int row = tile * 16 + n;
    int rowc = (row < V) ? row : (V - 1);   // clamp reads; pad rows never read back
    float rc = rcnt[rowc];
    const float* xr = X1 + ((long)m * Vp + rowc) * 24;
    v8f acc = {0.f, 0.f, 0.f, 0.f, 0.f, 0.f, 0.f, 0.f};
    #pragma unroll
    for (int kc = 0; kc < 6; ++kc) {
        int k0 = kc * 4;
        int ka = k0 + half * 2;
        float s = (ka >= 12) ? rc : 1.f;    // neighbor-mean columns get 1/cnt
        v2f a; a.x = xr[ka] * s; a.y = xr[ka + 1] * s;
        v2f w; w.x = Wc1[ka * 16 + n]; w.y = Wc1[(ka + 1) * 16 + n];
        acc = wmma4(a, w, acc);
    }
    float bn = b1[n];
    #pragma unroll
    for (int r = 0; r < 8; ++r) {
        int vtx = tile * 16 + r + half * 8;   // always < Vp: unconditional store
        float h = acc[r] + bn;
        h = (h > 0.f) ? h : 0.f;
        X2[((long)m * Vp + vtx) * 32 + n] = h;
    }
}

// --- MLP layer 2 + output head + state update, fused:
//   h2 = relu(X2(32) @ Wc2(32x16) + b2)      (WMMA, 8 K-chunks)
//   dV = h2 @ Wo16 + bo                      (WMMA via LDS transpose)
//   v += STEP*dV ; sulc += STEP*(n . dV)
__global__ void k_mlp2(const float* __restrict__ X2, const float* __restrict__ Wc2,
                       const float* __restrict__ b2, const float* __restrict__ Wo16,
                       const float* __restrict__ bo, const float* __restrict__ rcnt,
                       const float* __restrict__ nrm, float* __restrict__ vcur,
                       float* __restrict__ sulc, int V, int Vp) {
    __shared__ float hsm[8 * 256];
    int m = blockIdx.y;
    int wid = threadIdx.x >> 5;
    int lane = threadIdx.x & 31;
    int tile = blockIdx.x * 8 + wid;
    int n = lane & 15;
    int half = lane >> 4;
    int row = tile * 16 + n;
    int rowc = (row < V) ? row : (V - 1);
    float rc = rcnt[rowc];
    const float* xr = X2 + ((long)m * Vp + rowc) * 32;
    v8f acc = {0.f, 0.f, 0.f, 0.f, 0.f, 0.f, 0.f, 0.f};
    #pragma unroll
    for (int kc = 0; kc < 8; ++kc) {
        int k0 = kc * 4;
        int ka = k0 + half * 2;
        float s = (ka >= 16) ? rc : 1.f;
        v2f a; a.x = xr[ka] * s; a.y = xr[ka + 1] * s;
        v2f w; w.x = Wc2[ka * 16 + n]; w.y = Wc2[(ka + 1) * 16 + n];
        acc = wmma4(a, w, acc);
    }
    float bn = b2[n];
    float* hw = hsm + wid * 256;
    #pragma unroll
    for (int r = 0; r < 8; ++r) {
        float h = acc[r] + bn;
        h = (h > 0.f) ? h : 0.f;
        hw[(r + half * 8) * 16 + n] = h;  // D-layout -> row-major in LDS
    }
    __syncthreads();
    v8f accd = {0.f, 0.f, 0.f, 0.f, 0.f, 0.f, 0.f, 0.f};
    #pragma unroll
    for (int kc = 0; kc < 4; ++kc) {
        int k0 = kc * 4;
        int ka = k0 + half * 2;
        v2f a; a.x = hw[n * 16 + ka]; a.y = hw[n * 16 + ka + 1];  // A-layout reload
        v2f w; w.x = Wo16[ka * 16 + n]; w.y = Wo16[(ka + 1) * 16 + n];
        accd = wmma4(a, w, accd);
    }
    if (n < 3) {
        float bc = bo[n];
        #pragma unroll
        for (int r = 0; r < 8; ++r) {
            int vtx = tile * 16 + r + half * 8;   // < Vp: pad rows are write-only
            long idx = (long)m * Vp + vtx;
            float dv = accd[r] + bc;
            vcur[idx * 3 + n] += STEPF * dv;
            atomicAdd(&sulc[idx], STEPF * nrm[idx * 3 + n] * dv);
        }
    }
}

// -------------------------------------------------- final output assembly
__global__ void k_finalize(const float* __restrict__ vcur, const float* __restrict__ sulc,
                           const float* __restrict__ cs, float* __restrict__ out,
                           int V, int Vp) {
    long tid = (long)blockIdx.x * blockDim.x + threadIdx.x;
    long total = (long)NM * V;
    if (tid >= total) return;
    int m = (int)(tid / V), i = (int)(tid % V);
    long idx = (long)m * Vp + i;
    float* o = out + tid * 4;
    o[0] = vcur[idx * 3 + 0] * cs[m * 4 + 0];
    o[1] = vcur[idx * 3 + 1] * cs[m * 4 + 1];
    o[2] = vcur[idx * 3 + 2] * cs[m * 4 + 2];
    o[3] = sulc[idx];
}

extern "C" void kernel_launch(void* const* d_in, const int* in_sizes, int n_in,
                              void* d_out, int out_size, void* d_ws, size_t ws_size,
                              hipStream_t stream) {
    (void)n_in; (void)out_size; (void)ws_size;
    const float* vlh = (const float*)d_in[0];
    const float* vrh = (const float*)d_in[1];
    const float* W1s = (const float*)d_in[2];
    const float* W1n = (const float*)d_in[3];
    const float* b1  = (const float*)d_in[4];
    const float* W2s = (const float*)d_in[5];
    const float* W2n = (const float*)d_in[6];
    const float* b2  = (const float*)d_in[7];
    const float* Wo  = (const float*)d_in[8];
    const float* bo  = (const float*)d_in[9];
    const int* faces = (const int*)d_in[10];
    int V = in_sizes[0] / (NB * 3);
    int F = in_sizes[10] / 3;
    int tiles = (V + 15) / 16;
    int tiles8 = ((tiles + 7) / 8) * 8;     // round tiles to whole 8-wave blocks
    int Vp = tiles8 * 16;                   // padded per-mesh vertex stride

    float* ws = (float*)d_ws;
    long off = 0;
    auto alloc = [&](long nelem) {          // 256B-aligned sub-allocations
        float* p = ws + off;
        off += (nelem + 63) & ~63L;
        return p;
    };
    float* vcur = alloc((long)NM * Vp * 3);
    float* sulc = alloc((long)NM * Vp);
    float* cs   = alloc(NM * 4);
    float* cnt  = alloc(V);
    float* rcnt = alloc(V);
    float* Wc1  = alloc(24 * 16);
    float* Wc2  = alloc(32 * 16);
    float* Wo16 = alloc(16 * 16);
    float* nrm  = alloc((long)NM * Vp * 3);
    float* lap  = alloc((long)NM * Vp * 3);
    float* X1   = alloc((long)NM * Vp * 24);
    float* X2   = alloc((long)NM * Vp * 32);
    float* zbase = nrm;                     // nrm|lap|X1|X2 (plus align pads)
    long zcount = (ws + off) - nrm;

    k_zero<<<256, 256, 0, stream>>>(cnt, (long)V);
    k_count_faces<<<(F + 255) / 256, 256, 0, stream>>>(faces, cnt, F);
    k_recip<<<(V + 255) / 256, 256, 0, stream>>>(cnt, rcnt, V);
    k_prep_weights<<<1, 512, 0, stream>>>(W1s, W1n, W2s, W2n, Wo, Wc1, Wc2, Wo16);
    k_prep_mesh<<<NM, 256, 0, stream>>>(vlh, vrh, vcur, sulc, cs, V, Vp);

    long nfaceT = (long)NM * F;
    int fgBlocks = (int)((nfaceT + 255) / 256);
    long nvertT = (long)NM * V;
    int vBlocks = (int)((nvertT + 255) / 256);
    dim3 mlpGrid(tiles8 / 8, NM);

    for (int s = 0; s < 10; ++s) {
        k_zero<<<4096, 256, 0, stream>>>(zbase, zcount);
        k_face_geom<<<fgBlocks, 256, 0, stream>>>(faces, vcur, nrm, lap, Vp, F);
        k_vert_feats<<<vBlocks, 256, 0, stream>>>(vcur, nrm, lap, X1, V, Vp);
        k_nbr_scatter<24, 9, 12><<<fgBlocks, 256, 0, stream>>>(faces, X1, Vp, F);
        k_mlp1<<<mlpGrid, 256, 0, stream>>>(X1, Wc1, b1, rcnt, X2, V, Vp);
        k_nbr_scatter<32, 16, 16><<<fgBlocks, 256, 0, stream>>>(faces, X2, Vp, F);
        k_mlp2<<<mlpGrid, 256, 0, stream>>>(X2, Wc2, b2, Wo16, bo, rcnt, nrm, vcur, sulc, V, Vp);
    }
    k_finalize<<<vBlocks, 256, 0, stream>>>(vcur, sulc, cs, (float*)d_out, V, Vp);
}